// GCN_66941360276307
// MI455X (gfx1250) — compile-verified
//
#include <hip/hip_runtime.h>
#include <stdint.h>

// ---------------------------------------------------------------------------
// GCN on MI455X (gfx1250): CSR build -> gather-aggregate (no float atomics)
// -> fp32 WMMA GEMM (v_wmma_f32_16x16x4_f32) with LDS-staged A tile and
// fused bias/PReLU/pooling epilogue.
// ---------------------------------------------------------------------------

typedef __attribute__((ext_vector_type(2))) float v2f;
typedef __attribute__((ext_vector_type(8))) float v8f;

#define DIM 256          // IN_DIM == OUT_DIM == 256
#define DIM_W 8          // floats per lane per row (256 / 32 lanes)
#define A_PITCH 260      // LDS row pitch (words): bank-conflict-free for ds_load_b64

// ---------------- zero helper ----------------
__global__ void gcn_zero_i32(int* __restrict__ p, int n) {
    int i = blockIdx.x * blockDim.x + threadIdx.x;
    if (i < n) p[i] = 0;
}

// ---------------- degree histogram (int atomics, 2 per edge) ----------------
__global__ void gcn_degrees(const long long* __restrict__ src,
                            const long long* __restrict__ dst,
                            int* __restrict__ outdeg, int* __restrict__ indeg, int E) {
    int i = blockIdx.x * blockDim.x + threadIdx.x;
    if (i < E) {
        atomicAdd(&outdeg[(int)src[i]], 1);
        atomicAdd(&indeg[(int)dst[i]], 1);
    }
}

// ---------------- norms: rsqrt(max(deg,1)) ----------------
__global__ void gcn_norms(const int* __restrict__ outdeg, const int* __restrict__ indeg,
                          float* __restrict__ norm_s, float* __restrict__ norm_d, int N) {
    int i = blockIdx.x * blockDim.x + threadIdx.x;
    if (i < N) {
        norm_s[i] = rsqrtf(fmaxf((float)outdeg[i], 1.0f));
        norm_d[i] = rsqrtf(fmaxf((float)indeg[i], 1.0f));
    }
}

// ---------------- single-block exclusive scan over indeg -> csr_off ----------------
__global__ void __launch_bounds__(1024)
gcn_scan(const int* __restrict__ indeg, int* __restrict__ off,
         int* __restrict__ cur, int N) {
    __shared__ int buf[1024];
    __shared__ int carry;
    if (threadIdx.x == 0) carry = 0;
    __syncthreads();
    for (int base = 0; base < N; base += 1024) {
        int i = base + (int)threadIdx.x;
        int v = (i < N) ? indeg[i] : 0;
        buf[threadIdx.x] = v;
        __syncthreads();
        // Hillis-Steele inclusive scan in LDS
        for (int ofs = 1; ofs < 1024; ofs <<= 1) {
            int t = (threadIdx.x >= (unsigned)ofs) ? buf[threadIdx.x - ofs] : 0;
            __syncthreads();
            buf[threadIdx.x] += t;
            __syncthreads();
        }
        int excl = buf[threadIdx.x] - v + carry;   // reads old carry
        if (i < N) { off[i] = excl; cur[i] = excl; }
        __syncthreads();
        if (threadIdx.x == 1023) carry += buf[1023];
        __syncthreads();
    }
    if (threadIdx.x == 0) off[N] = carry;   // == E
}

// ---------------- scatter edges into CSR (int atomic per edge) ----------------
__global__ void gcn_scatter(const long long* __restrict__ src,
                            const long long* __restrict__ dst,
                            int* __restrict__ cur, int* __restrict__ csr_src, int E) {
    int i = blockIdx.x * blockDim.x + threadIdx.x;
    if (i < E) {
        int pos = atomicAdd(&cur[(int)dst[i]], 1);
        csr_src[pos] = (int)src[i];
    }
}

// ---------------- aggregation: one wave32 per destination node ----------------
// agg[d,:] = norm_d[d] * sum_{e: dst==d} x[csr_src[e],:] * norm_s[csr_src[e]]
__global__ void __launch_bounds__(256)
gcn_aggregate(const float* __restrict__ x, const int* __restrict__ off,
              const int* __restrict__ csr_src,
              const float* __restrict__ norm_s, const float* __restrict__ norm_d,
              float* __restrict__ agg, int N) {
    int node = blockIdx.x * (blockDim.x >> 5) + (threadIdx.x >> 5);
    int lane = threadIdx.x & 31;
    if (node >= N) return;
    float acc[DIM_W];
#pragma unroll
    for (int j = 0; j < DIM_W; ++j) acc[j] = 0.0f;

    int e0 = off[node], e1 = off[node + 1];
    for (int e = e0; e < e1; ++e) {
        int s = csr_src[e];
        float ns = norm_s[s];
        const float* row = x + (size_t)s * DIM;
        if (e + 1 < e1) {
            // prefetch next gathered row (one cacheline per low lane) -> global_prefetch_b8
            int s2 = csr_src[e + 1];
            __builtin_prefetch(x + (size_t)s2 * DIM + (lane & 7) * 32, 0, 1);
        }
#pragma unroll
        for (int j = 0; j < DIM_W; ++j)
            acc[j] += row[j * 32 + lane] * ns;
    }
    float nd = norm_d[node];
    float* o = agg + (size_t)node * DIM;
#pragma unroll
    for (int j = 0; j < DIM_W; ++j)
        o[j * 32 + lane] = acc[j] * nd;
}

// ---------------- WMMA GEMM + bias + PReLU + column-sum pooling ----------------
// out[16*blk + m, :] = prelu(agg_tile @ W + b);  hg[n] += column sums.
// 512 threads = 16 waves; wave w owns N-columns [16w, 16w+16). A tile (16x256)
// staged once in LDS (padded pitch 260 -> ds_load_b64 conflict-free); K looped
// in steps of 4 via V_WMMA_F32_16X16X4_F32 (exact fp32 matrix math).
__global__ void __launch_bounds__(512)
gcn_gemm_prelu_pool(const float* __restrict__ agg, const float* __restrict__ W,
                    const float* __restrict__ bias, const float* __restrict__ prelu_a,
                    float* __restrict__ out, float* __restrict__ hg) {
    __shared__ float As[16 * A_PITCH];          // 16.3 KB

    const int tid   = threadIdx.x;
    const int lane  = tid & 31;
    const int wv    = tid >> 5;                 // 0..15 : N-tile
    const int mbase = blockIdx.x * 16;          // M-tile (50000 % 16 == 0)
    const int half  = lane >> 4;                // 0|1
    const int m     = lane & 15;                // A row in tile / B,C column in tile
    const int ncol  = (wv << 4) + m;            // global output column

    // Cooperative A-tile stage: 1024 float4s, 2 per thread (b128 global + b128 ds)
    const float4* gsrc = (const float4*)(agg + (size_t)mbase * DIM);
#pragma unroll
    for (int i = 0; i < 2; ++i) {
        int idx = tid + i * 512;                // float4 index 0..1023
        int r   = idx >> 6;                     // 64 float4s per 256-float row
        int c   = (idx & 63) << 2;              // float column
        *(float4*)&As[r * A_PITCH + c] = gsrc[idx];
    }
    __syncthreads();

    // A fragment: row m, K pair starting at 2*half (LDS)
    const float* arow = As + m * A_PITCH + 2 * half;
    // B fragment: W[k][n] row-major, rows 2*half and 2*half+1 of each K-step
    const float* bcol = W + (size_t)(2 * half) * DIM + ncol;

    v8f c = {};
#pragma unroll 4
    for (int k0 = 0; k0 < DIM; k0 += 4) {
        v2f a, b;
        a.x = arow[k0];
        a.y = arow[k0 + 1];
        b.x = bcol[k0 * DIM];
        b.y = bcol[(k0 + 1) * DIM];
        c = __builtin_amdgcn_wmma_f32_16x16x4_f32(
                /*neg_a=*/false, a, /*neg_b=*/false, b,
                /*c_mod=*/(short)0, c, /*reuse_a=*/false, /*reuse_b=*/false);
    }

    const float bb = bias[ncol];
    const float al = *prelu_a;
    float colsum = 0.0f;
#pragma unroll
    for (int j = 0; j < 8; ++j) {
        float v = c[j] + bb;
        v = (v >= 0.0f) ? v : al * v;
        out[(size_t)(mbase + half * 8 + j) * DIM + ncol] = v;
        colsum += v;
    }
    atomicAdd(&hg[ncol], colsum);   // lanes l and l+16 share ncol; both add (disjoint rows)
}

// ---------------------------------------------------------------------------
extern "C" void kernel_launch(void* const* d_in, const int* in_sizes, int n_in,
                              void* d_out, int out_size, void* d_ws, size_t ws_size,
                              hipStream_t stream) {
    const float*     feat    = (const float*)d_in[0];
    const long long* src     = (const long long*)d_in[1];   // jnp.int64
    const long long* dst     = (const long long*)d_in[2];
    const float*     W0      = (const float*)d_in[3];
    const float*     b0      = (const float*)d_in[4];
    const float*     W1      = (const float*)d_in[5];
    const float*     b1      = (const float*)d_in[6];
    const float*     prelu_a = (const float*)d_in[7];

    const int N = in_sizes[0] / DIM;   // 50000
    const int E = in_sizes[1];         // 800000

    // ---- workspace carve-out ----
    int*   outdeg  = (int*)d_ws;               // [N]
    int*   indeg   = outdeg + N;               // [N] (contiguous with outdeg)
    int*   csr_off = indeg + N;                // [N+1]
    int*   csr_cur = csr_off + N + 1;          // [N]
    int*   csr_src = csr_cur + N;              // [E]
    float* norm_s  = (float*)(csr_src + E);    // [N]
    float* norm_d  = norm_s + N;               // [N]
    float* agg     = (float*)(((uintptr_t)(norm_d + N) + 15) & ~(uintptr_t)15); // [N*256]
    (void)ws_size; (void)n_in; (void)out_size;

    float* h   = (float*)d_out;                 // [N, 256] final layer output
    float* hg  = h + (size_t)N * DIM;           // [512] pooled vector

    const int TPB = 256;
    // 1) zero degree counters (contiguous 2N ints) and hg (512 floats; 0 bits)
    gcn_zero_i32<<<(2 * N + TPB - 1) / TPB, TPB, 0, stream>>>(outdeg, 2 * N);
    gcn_zero_i32<<<(512 + TPB - 1) / TPB, TPB, 0, stream>>>((int*)hg, 512);
    // 2) degrees
    gcn_degrees<<<(E + TPB - 1) / TPB, TPB, 0, stream>>>(src, dst, outdeg, indeg, E);
    // 3) norms
    gcn_norms<<<(N + TPB - 1) / TPB, TPB, 0, stream>>>(outdeg, indeg, norm_s, norm_d, N);
    // 4) CSR offsets (single-block scan) + scatter
    gcn_scan<<<1, 1024, 0, stream>>>(indeg, csr_off, csr_cur, N);
    gcn_scatter<<<(E + TPB - 1) / TPB, TPB, 0, stream>>>(src, dst, csr_cur, csr_src, E);

    const int aggBlocks  = (N + 7) / 8;   // 8 waves per 256-thread block, 1 node/wave
    const int gemmBlocks = N / 16;        // 3125 M-tiles, exact

    // ---- layer 1: feat -> agg -> h (stored in d_out h-region), hg[0:256] ----
    gcn_aggregate<<<aggBlocks, TPB, 0, stream>>>(feat, csr_off, csr_src,
                                                 norm_s, norm_d, agg, N);
    gcn_gemm_prelu_pool<<<gemmBlocks, 512, 0, stream>>>(agg, W0, b0, prelu_a, h, hg);

    // ---- layer 2: h -> agg (overwrite) -> h (overwrite; GEMM reads only agg), hg[256:512]
    gcn_aggregate<<<aggBlocks, TPB, 0, stream>>>(h, csr_off, csr_src,
                                                 norm_s, norm_d, agg, N);
    gcn_gemm_prelu_pool<<<gemmBlocks, 512, 0, stream>>>(agg, W1, b1, prelu_a, h, hg + DIM);
}